// Similarity_40527311405617
// MI455X (gfx1250) — compile-verified
//
#include <hip/hip_runtime.h>
#include <hip/hip_bf16.h>

// ---- problem constants (from reference) ----
#define BN   256   // B*N
#define Lr   64    // L
#define Dd   96    // D
#define LDSS 100   // padded LDS row stride (floats): 100*4 % 16 == 0, 100%64=36 -> conflict-free

typedef __attribute__((ext_vector_type(2))) float v2f;
typedef __attribute__((ext_vector_type(4))) float f4;
typedef __attribute__((ext_vector_type(8))) float v8f;

#if __has_builtin(__builtin_amdgcn_tanhf)
#define TANHF(v) __builtin_amdgcn_tanhf(v)
#else
#define TANHF(v) tanhf(v)
#endif

__global__ __launch_bounds__(256) void sim_fused_kernel(
    const float* __restrict__ x,     // (BN, L, D)
    const float* __restrict__ FC1,   // (D, D)
    const float* __restrict__ FC2,   // (D, D)
    const float* __restrict__ att,   // (D, 1)
    float* __restrict__ out)         // (BN, L, L)
{
    __shared__ float xs[Lr * LDSS];   // x tile
    __shared__ float As[Lr * LDSS];   // x @ FC1
    __shared__ float Cs[Lr * LDSS];   // x @ FC2
    __shared__ float attS[Dd];

    const int bn = blockIdx.x;                 // 0..255
    const int t  = threadIdx.x;                // 0..255
    const float* xb = x + (size_t)bn * Lr * Dd;

    // -------- stage x (64x96) into LDS, padded stride, float4 --------
    for (int idx = t; idx < Lr * (Dd / 4); idx += 256) {
        const int row = idx / (Dd / 4);
        const int c4  = (idx % (Dd / 4)) * 4;
        f4 v = *(const f4*)(xb + row * Dd + c4);
        *(f4*)(&xs[row * LDSS + c4]) = v;
    }
    if (t < Dd) attS[t] = att[t];
    __syncthreads();

    // -------- GEMMs via v_wmma_f32_16x16x4_f32 --------
    // waves 0..3: A = x@FC1 (mtile = wave); waves 4..7: C = x@FC2 (mtile = wave-4)
    const int wave  = t >> 5;
    const int lane  = t & 31;
    const int mtile = wave & 3;
    const float* W  = (wave < 4) ? FC1 : FC2;
    float*     Dst  = (wave < 4) ? As  : Cs;

    const int mrow = mtile * 16 + (lane & 15);   // A-frag: row per lane
    const int kk   = (lane >> 4) * 2;            // K sub-offset per half-wave

    for (int nt = 0; nt < 6; ++nt) {
        const int n = nt * 16 + (lane & 15);     // B-frag / D column per lane
        v8f acc = {};
        #pragma unroll
        for (int k4 = 0; k4 < Dd; k4 += 4) {
            // A-matrix 16x4 f32 layout: v0 = A[m][k4+kk], v1 = A[m][k4+kk+1]
            v2f a = *(const v2f*)(&xs[mrow * LDSS + k4 + kk]);
            // B-matrix 4x16 f32 layout (rows striped across lanes)
            v2f b;
            b.x = W[(k4 + kk)     * Dd + n];
            b.y = W[(k4 + kk + 1) * Dd + n];
            acc = __builtin_amdgcn_wmma_f32_16x16x4_f32(
                /*neg_a=*/false, a, /*neg_b=*/false, b,
                /*c_mod=*/(short)0, acc, /*reuse_a=*/false, /*reuse_b=*/false);
        }
        // D layout: VGPR r holds M = r + 8*(lane>>4), N = n
        const int drow0 = mtile * 16 + 8 * (lane >> 4);
        #pragma unroll
        for (int r = 0; r < 8; ++r)
            Dst[(drow0 + r) * LDSS + n] = acc[r];
    }
    __syncthreads();

    // -------- fused epilogue: sim[i,j] = sum_d tanh(A[j,d]+C[i,d]) * att[d] --------
    // thread t owns 4x4 tile: i in [i0,i0+4), j in [j0,j0+4)
    const int i0 = (t >> 4) * 4;
    const int j0 = (t & 15) * 4;

    float acc[4][4] = {};
    for (int d = 0; d < Dd; d += 4) {
        f4 w4 = *(const f4*)(&attS[d]);
        f4 ad[4], cd[4];
        #pragma unroll
        for (int u = 0; u < 4; ++u) {
            ad[u] = *(const f4*)(&As[(j0 + u) * LDSS + d]);
            cd[u] = *(const f4*)(&Cs[(i0 + u) * LDSS + d]);
        }
        #pragma unroll
        for (int ii = 0; ii < 4; ++ii) {
            #pragma unroll
            for (int jj = 0; jj < 4; ++jj) {
                #pragma unroll
                for (int c = 0; c < 4; ++c)
                    acc[ii][jj] += TANHF(ad[jj][c] + cd[ii][c]) * w4[c];
            }
        }
    }

    float* ob = out + (size_t)bn * Lr * Lr;
    #pragma unroll
    for (int ii = 0; ii < 4; ++ii) {
        f4 v;
        v.x = acc[ii][0]; v.y = acc[ii][1]; v.z = acc[ii][2]; v.w = acc[ii][3];
        *(f4*)(&ob[(i0 + ii) * Lr + j0]) = v;
    }
}

extern "C" void kernel_launch(void* const* d_in, const int* in_sizes, int n_in,
                              void* d_out, int out_size, void* d_ws, size_t ws_size,
                              hipStream_t stream) {
    const float* x   = (const float*)d_in[0];   // (2,128,64,96)
    const float* FC1 = (const float*)d_in[1];   // (96,96)
    const float* FC2 = (const float*)d_in[2];   // (96,96)
    const float* att = (const float*)d_in[3];   // (96,1)
    float* out = (float*)d_out;                 // (2,128,64,64)

    sim_fused_kernel<<<dim3(BN), dim3(256), 0, stream>>>(x, FC1, FC2, att, out);
}